// Predictor_283467842752
// MI455X (gfx1250) — compile-verified
//
#include <hip/hip_runtime.h>
#include <stdint.h>

typedef __attribute__((ext_vector_type(2))) float v2f;
typedef __attribute__((ext_vector_type(8))) float v8f;

#define DIM       1024
#define NBLK      64          // DIM / 16 column blocks
#define MTILE     32          // batch rows per workgroup
#define OT_STRIDE 1028        // padded LDS row stride (1028 % 64 == 4 -> conflict free,
                              //  and 1028*4 B is 16B aligned for async B128 copies)
#define NTHREADS  256
#define NWAVES    8

// Forward substitution over the causal graph:
//   O[:, i] = O @ W[i] + noise[:, i]   (W strictly lower triangular)
//   O[:, target] = value
// Blocked by 16 columns. Inter-block coupling = GEMM via V_WMMA_F32_16X16X4_F32
// (K strided over 8 waves). Intra-block = 16-step sequential triangular sweep.
// The W row-panel for block b+1 is prefetched into a double-buffered LDS slot
// with GLOBAL_LOAD_ASYNC_TO_LDS_B128 while block b's GEMM runs (ASYNCcnt).
__global__ __launch_bounds__(NTHREADS)
void fwd_subst_wmma_kernel(const float* __restrict__ noise,
                           const float* __restrict__ W,
                           const float* __restrict__ valuep,
                           const int*   __restrict__ targetp,
                           float*       __restrict__ out) {
  extern __shared__ float smem[];
  float* Ot  = smem;                          // [MTILE][OT_STRIDE] output panel
  float* Wp0 = Ot + MTILE * OT_STRIDE;        // [16][OT_STRIDE] W panel, buffer 0
  float* Wp1 = Wp0 + 16 * OT_STRIDE;          // [16][OT_STRIDE] W panel, buffer 1
  float* Pr  = Wp1 + 16 * OT_STRIDE;          // [NWAVES][MTILE][16] partial accs
  float* Cb  = Pr + NWAVES * MTILE * 16;      // [MTILE][17] current block
  float* Wd  = Cb + MTILE * 17;               // [16][17]    diagonal W block

  const int t     = threadIdx.x;
  const int wave  = __builtin_amdgcn_readfirstlane(t >> 5);  // scalar wave id
  const int lane  = t & 31;
  const int lhalf = lane >> 4;                // 0: K+0/1, 1: K+2/3 operand split
  const int l15   = lane & 15;
  const int row0  = blockIdx.x * MTILE;

  const float value  = valuep[0];
  const int   target = targetp[0];

  for (int b = 0; b < NBLK; ++b) {
    const int bcol = b * 16;
    float* Wp  = (b & 1) ? Wp1 : Wp0;         // panel for THIS block (prefetched)
    float* Wpn = (b & 1) ? Wp0 : Wp1;         // panel for NEXT block

    // ---- stage 16x16 diagonal block (regular loads; tiny) ------------------
    {
      const int i = t >> 4, j = t & 15;
      Wd[i * 17 + j] = W[(size_t)(bcol + i) * DIM + bcol + j];
    }

    // Retire this wave's async panel loads issued last block, then rendezvous:
    // after the barrier every wave sees the full Wp panel and prev writeback.
    asm volatile("s_wait_asynccnt 0x0" ::: "memory");
    __syncthreads();

    // ---- async-prefetch next W panel: rows [bcol+16, bcol+32), cols [0, bcol+16)
    if (b + 1 < NBLK) {
      const int chunks = 4 * (b + 1);         // 16B chunks per row (<= 252)
      if (t < chunks) {
#pragma unroll 1
        for (int i = 0; i < 16; ++i) {
          const float* g = W + (size_t)(bcol + 16 + i) * DIM + 4 * t;
          const uint32_t lds = (uint32_t)(uintptr_t)(Wpn + i * OT_STRIDE + 4 * t);
          asm volatile("global_load_async_to_lds_b128 %0, %1, off"
                       :: "v"(lds), "v"(g) : "memory");
        }
      }
      // prefetch next block's noise tile (one 64B row segment per lane)
      if (t < MTILE)
        __builtin_prefetch(noise + (size_t)(row0 + t) * DIM + bcol + 16, 0, 0);
    }

    // ---- GEMM: acc[m][n] = sum_{k < bcol} Ot[m][k] * W[bcol+n][k] ----------
    v8f acc0 = {0.f, 0.f, 0.f, 0.f, 0.f, 0.f, 0.f, 0.f};
    v8f acc1 = {0.f, 0.f, 0.f, 0.f, 0.f, 0.f, 0.f, 0.f};
#pragma unroll 2
    for (int k = wave * 4; k < bcol; k += NWAVES * 4) {
      const int ka = k + lhalf * 2;           // per-lane K pair
      v2f bmat;                               // B[k][n] = W[bcol+n][k]
      bmat.x = Wp[l15 * OT_STRIDE + ka];
      bmat.y = Wp[l15 * OT_STRIDE + ka + 1];
      v2f a0, a1;                             // A[m][k] = Ot[m][k]
      a0.x = Ot[l15 * OT_STRIDE + ka];
      a0.y = Ot[l15 * OT_STRIDE + ka + 1];
      a1.x = Ot[(l15 + 16) * OT_STRIDE + ka];
      a1.y = Ot[(l15 + 16) * OT_STRIDE + ka + 1];
      acc0 = __builtin_amdgcn_wmma_f32_16x16x4_f32(false, a0, false, bmat,
                                                   (short)0, acc0, false, false);
      acc1 = __builtin_amdgcn_wmma_f32_16x16x4_f32(false, a1, false, bmat,
                                                   (short)0, acc1, false, false);
    }
    // C/D layout: VGPR r -> (M = r + 8*lhalf, N = l15)
#pragma unroll
    for (int r = 0; r < 8; ++r) {
      Pr[(wave * MTILE + (r + 8 * lhalf)) * 16 + l15]      = acc0[r];
      Pr[(wave * MTILE + (r + 8 * lhalf + 16)) * 16 + l15] = acc1[r];
    }
    __syncthreads();

    // ---- reduce 8 partials + noise into current block staging --------------
    for (int e = t; e < MTILE * 16; e += NTHREADS) {
      const int m = e >> 4, n = e & 15;
      float s = noise[(size_t)(row0 + m) * DIM + bcol + n];
#pragma unroll
      for (int w = 0; w < NWAVES; ++w)
        s += Pr[(w * MTILE + m) * 16 + n];
      Cb[m * 17 + n] = s;
    }
    __syncthreads();

    // ---- intra-block sequential forward substitution (16 steps) ------------
    if (t < MTILE) {
      const int m = t;
#pragma unroll 1
      for (int i = 0; i < 16; ++i) {
        float s = Cb[m * 17 + i];
        for (int j = 0; j < i; ++j)
          s += Cb[m * 17 + j] * Wd[i * 17 + j];
        if (bcol + i == target) s = value;    // do-intervention
        Cb[m * 17 + i] = s;
      }
    }
    __syncthreads();

    // ---- writeback: LDS output panel + global output -----------------------
    for (int e = t; e < MTILE * 16; e += NTHREADS) {
      const int m = e >> 4, n = e & 15;
      const float s = Cb[m * 17 + n];
      Ot[m * OT_STRIDE + bcol + n] = s;
      out[(size_t)(row0 + m) * DIM + bcol + n] = s;
    }
    __syncthreads();
  }
}

extern "C" void kernel_launch(void* const* d_in, const int* in_sizes, int n_in,
                              void* d_out, int out_size, void* d_ws, size_t ws_size,
                              hipStream_t stream) {
  const float* noise   = (const float*)d_in[0];
  const float* weights = (const float*)d_in[1];
  const float* value   = (const float*)d_in[2];
  const int*   target  = (const int*)d_in[3];
  float*       out     = (float*)d_out;

  const int batch = in_sizes[0] / DIM;        // 16384
  dim3 grid(batch / MTILE);                   // 512 workgroups
  dim3 block(NTHREADS);                       // 8 waves (wave32)
  const size_t smem = (size_t)(MTILE * OT_STRIDE + 2 * 16 * OT_STRIDE +
                               NWAVES * MTILE * 16 + MTILE * 17 + 16 * 17) *
                      sizeof(float);          // ~276 KB of the 320 KB WGP LDS
  hipLaunchKernelGGL(fwd_subst_wmma_kernel, grid, block, smem, stream,
                     noise, weights, value, target, out);
}